// JacobianAnalyzer_30794915512670
// MI455X (gfx1250) — compile-verified
//
#include <hip/hip_runtime.h>
#include <math.h>

typedef float v2f __attribute__((ext_vector_type(2)));
typedef float v8f __attribute__((ext_vector_type(8)));
typedef unsigned int v4u __attribute__((ext_vector_type(4)));
typedef int v4i __attribute__((ext_vector_type(4)));
typedef int v8i __attribute__((ext_vector_type(8)));

#define Bsz  64
#define NSEQ 192
#define Lh   48
#define Dh   128
#define PRE  32
#define SUF  16

// workspace layout (float offsets)
#define X_OFF   ((size_t)0)
#define X_LEN   ((size_t)NSEQ * Lh * Dh)            // 1,179,648
#define H_OFF   (X_OFF + X_LEN)
#define H_LEN   ((size_t)NSEQ * Dh)
#define HF_OFF  (H_OFF + H_LEN)
#define HF_LEN  ((size_t)NSEQ * Dh)
#define J_OFF   (HF_OFF + HF_LEN)
#define J_LEN   ((size_t)NSEQ * Dh * Dh)            // 3,145,728
#define SV_OFF  (J_OFF + J_LEN)
#define SV_LEN  ((size_t)NSEQ * Dh)
#define M_OFF   (SV_OFF + SV_LEN)
#define M_LEN   ((size_t)Bsz * 2 * Dh * Dh)         // 2,097,152
#define MSV_OFF (M_OFF + M_LEN)
#define MSV_LEN ((size_t)Bsz * 2 * Dh)

#define RANK_THRESH (128.0f * 1.1920929e-7f)

// ---------------------------------------------------------------------------
// Tensor Data Mover: 2D tile (rows x cols) of f32, global <-> LDS.
// D# layout per CDNA5 ISA:
//  group0: [1:0]=count=1, [63:32]=lds_addr, [120:64]=global_addr, [127:126]=2
//  group1: [17:16]=data_size(2 -> 4B), [79:48]=tensor_dim0, [111:80]=tensor_dim1,
//          [127:112]=tile_dim0, [143:128]=tile_dim1, [207:160]=dim0_stride
//  groups 2/3 unused for 2D tiles (zero-filled).
// 6-arg builtin (amdgpu-toolchain / clang-23 lane): trailing zero v8i + cpol.
// ---------------------------------------------------------------------------
__device__ __forceinline__ void tdm_desc(v4u& g0, v8i& g1, void* lds, const void* gmem,
                                         int rows, int cols, int stride_elems) {
  unsigned long long ga = (unsigned long long)(size_t)gmem;
  g0[0] = 1u;                                           // count=1, is_restore=0
  g0[1] = (unsigned int)(size_t)lds;                    // lds_addr (byte offset)
  g0[2] = (unsigned int)ga;                             // global_addr[31:0]
  g0[3] = (unsigned int)((ga >> 32) & 0x01FFFFFFull) | (2u << 30);  // addr[56:32], type=2
  g1[0] = (int)(2u << 16);                              // data_size = 4 bytes
  g1[1] = (int)((unsigned)(cols & 0xFFFF) << 16);                       // dim0[15:0]
  g1[2] = (int)(((unsigned)(rows & 0xFFFF) << 16) | ((unsigned)cols >> 16)); // dim1[15:0] | dim0[31:16]
  g1[3] = (int)(((unsigned)(cols & 0xFFFF) << 16) | ((unsigned)rows >> 16)); // tile_dim0 | dim1[31:16]
  g1[4] = (int)(rows & 0xFFFF);                         // tile_dim1 (tile_dim2 = 0)
  g1[5] = stride_elems;                                 // tensor_dim0_stride[31:0]
  g1[6] = 0;
  g1[7] = 0;
}
__device__ __forceinline__ void tdm_load_2d_f32(void* lds, const void* gmem,
                                                int rows, int cols, int stride_elems) {
  v4u g0; v8i g1;
  tdm_desc(g0, g1, lds, gmem, rows, cols, stride_elems);
  v4i z4 = {0, 0, 0, 0};
  v8i z8 = {0, 0, 0, 0, 0, 0, 0, 0};
  __builtin_amdgcn_tensor_load_to_lds(g0, g1, z4, z4, z8, 0);
}
__device__ __forceinline__ void tdm_store_2d_f32(void* lds, void* gmem,
                                                 int rows, int cols, int stride_elems) {
  v4u g0; v8i g1;
  tdm_desc(g0, g1, lds, gmem, rows, cols, stride_elems);
  v4i z4 = {0, 0, 0, 0};
  v8i z8 = {0, 0, 0, 0, 0, 0, 0, 0};
  __builtin_amdgcn_tensor_store_from_lds(g0, g1, z4, z4, z8, 0);
}

// ---------------------------------------------------------------------------
// K1: x[seq][l][d] = embedding[token][d] + sinusoidal PE(l, d)
// ---------------------------------------------------------------------------
__global__ void embed_pe_kernel(const int* __restrict__ tok,
                                const float* __restrict__ emb,
                                float* __restrict__ x) {
  int idx = blockIdx.x * blockDim.x + threadIdx.x;
  if (idx >= NSEQ * Lh * Dh) return;
  int d = idx & (Dh - 1);
  int l = (idx / Dh) % Lh;
  int s = idx / (Dh * Lh);
  int t = tok[s * Lh + l];
  __builtin_prefetch(&emb[t * Dh], 0, 3);
  float f   = __expf(-(float)(2 * (d >> 1)) * (logf(10000.0f) / (float)Dh));
  float ang = (float)l * f;
  float pe  = (d & 1) ? __cosf(ang) : __sinf(ang);
  x[idx] = emb[t * Dh + d] + pe;
}

// ---------------------------------------------------------------------------
// K2: prefix RNN, 32 steps. 1 block = 16 sequences, 8 waves, W in LDS via TDM.
// Each wave owns one 16-col tile of h'. WMMA f32 16x16x4.
// ---------------------------------------------------------------------------
__global__ __launch_bounds__(256)
void prefix_kernel(const float* __restrict__ x,
                   const float* __restrict__ W_ih, const float* __restrict__ W_hh,
                   const float* __restrict__ b_ih, const float* __restrict__ b_hh,
                   float* __restrict__ Hout) {
  __shared__ float sWih[Dh * Dh];
  __shared__ float sWhh[Dh * Dh];
  __shared__ float sh[16 * Dh];
  __shared__ float sxt[16 * Dh];
  int tid = threadIdx.x;
  int wv = tid >> 5, lane = tid & 31;
  int seq0 = blockIdx.x * 16;
  if (tid == 0) {
    tdm_load_2d_f32(sWih, W_ih, Dh, Dh, Dh);   // async DMA, TENSORcnt
    tdm_load_2d_f32(sWhh, W_hh, Dh, Dh, Dh);
  }
  for (int i = tid; i < 16 * Dh; i += 256) sh[i] = 0.0f;
  __builtin_amdgcn_s_wait_tensorcnt(0);
  __syncthreads();

  int m = lane & 15;              // A-frag row / B-frag col within tile
  int khalf = (lane >> 4) * 2;    // K sub-offset per half-wave
  int n = (wv << 4) + m;          // global output channel of this wave tile
  float bias = b_ih[n] + b_hh[n];
  v8f zero = {0.f, 0.f, 0.f, 0.f, 0.f, 0.f, 0.f, 0.f};

  for (int t = 0; t < PRE; ++t) {
    for (int i = tid; i < 16 * Dh; i += 256) {
      int r = i >> 7, c = i & 127;
      sxt[i] = x[(size_t)(seq0 + r) * Lh * Dh + (size_t)t * Dh + c];
    }
    __syncthreads();
    v8f acc = zero;
    for (int k0 = 0; k0 < Dh; k0 += 4) {
      int k = k0 + khalf;
      v2f a, b;
      a[0] = sxt[m * Dh + k];     a[1] = sxt[m * Dh + k + 1];
      b[0] = sWih[n * Dh + k];    b[1] = sWih[n * Dh + k + 1];
      acc = __builtin_amdgcn_wmma_f32_16x16x4_f32(false, a, false, b, (short)0, acc, false, false);
      a[0] = sh[m * Dh + k];      a[1] = sh[m * Dh + k + 1];
      b[0] = sWhh[n * Dh + k];    b[1] = sWhh[n * Dh + k + 1];
      acc = __builtin_amdgcn_wmma_f32_16x16x4_f32(false, a, false, b, (short)0, acc, false, false);
    }
    __syncthreads();   // all reads of sh done before overwrite
    for (int r = 0; r < 8; ++r) {
      int mm = r + ((lane >> 4) << 3);
      sh[mm * Dh + n] = tanhf(acc[r] + bias);
    }
    __syncthreads();
  }
  for (int i = tid; i < 16 * Dh; i += 256) {
    int r = i >> 7, c = i & 127;
    Hout[(size_t)(seq0 + r) * Dh + c] = sh[r * Dh + c];
  }
}

// ---------------------------------------------------------------------------
// K3: suffix RNN with Jacobian chain J <- diag(1-h'^2) * (W_hh @ J).
// 1 block per sequence; W_hh (TDM) + J in LDS; 8 waves x 8 register tiles.
// J written back to global via TDM store.
// ---------------------------------------------------------------------------
__global__ __launch_bounds__(256)
void suffix_jac_kernel(const float* __restrict__ x,
                       const float* __restrict__ W_ih_g, const float* __restrict__ W_hh_g,
                       const float* __restrict__ b_ih, const float* __restrict__ b_hh,
                       const float* __restrict__ Hin,
                       float* __restrict__ HFout, float* __restrict__ Jout) {
  __shared__ float sWhh[Dh * Dh];
  __shared__ float sJ[Dh * Dh];
  __shared__ float sh[Dh];
  __shared__ float shn[Dh];
  __shared__ float sdiag[Dh];
  int tid = threadIdx.x;
  int wv = tid >> 5, lane = tid & 31;
  int seq = blockIdx.x;
  if (tid == 0) tdm_load_2d_f32(sWhh, W_hh_g, Dh, Dh, Dh);
  for (int i = tid; i < Dh * Dh; i += 256)
    sJ[i] = ((i >> 7) == (i & 127)) ? 1.0f : 0.0f;   // J = I
  if (tid < Dh) sh[tid] = Hin[(size_t)seq * Dh + tid];
  __builtin_amdgcn_s_wait_tensorcnt(0);
  __syncthreads();

  int ml = lane & 15;
  int khalf = (lane >> 4) * 2;
  int mg = (wv << 4) + ml;       // A-frag global row (W_hh row)
  v8f zero = {0.f, 0.f, 0.f, 0.f, 0.f, 0.f, 0.f, 0.f};

  for (int t = 0; t < SUF; ++t) {
    // ---- h update (GEMV by 128 threads) ----
    if (tid < Dh) {
      int i = tid;
      float pre = b_ih[i] + b_hh[i];
      const float* xt = x + (size_t)seq * Lh * Dh + (size_t)(PRE + t) * Dh;
      for (int kk = 0; kk < Dh; ++kk) {
        int k = (kk + i) & (Dh - 1);   // stagger to spread LDS banks
        pre += xt[k] * W_ih_g[i * Dh + k] + sh[k] * sWhh[i * Dh + k];
      }
      float hn = tanhf(pre);
      shn[i] = hn;
      sdiag[i] = 1.0f - hn * hn;
    }
    __syncthreads();
    // ---- J' = diag * (W_hh @ J), computed into registers ----
    v8f acc[8];
    for (int q = 0; q < 8; ++q) acc[q] = zero;
    for (int k0 = 0; k0 < Dh; k0 += 4) {
      int k = k0 + khalf;
      v2f a;
      a[0] = sWhh[mg * Dh + k];
      a[1] = sWhh[mg * Dh + k + 1];
      for (int nt = 0; nt < 8; ++nt) {
        v2f b;
        int ng = (nt << 4) + ml;
        b[0] = sJ[k * Dh + ng];
        b[1] = sJ[(k + 1) * Dh + ng];
        acc[nt] = __builtin_amdgcn_wmma_f32_16x16x4_f32(false, a, false, b, (short)0, acc[nt], false, false);
      }
    }
    __syncthreads();   // everyone done reading sJ
    for (int nt = 0; nt < 8; ++nt) {
      for (int r = 0; r < 8; ++r) {
        int mm = (wv << 4) + r + ((lane >> 4) << 3);
        int nn = (nt << 4) + (lane & 15);
        sJ[mm * Dh + nn] = sdiag[mm] * acc[nt][r];
      }
    }
    if (tid < Dh) sh[tid] = shn[tid];
    __syncthreads();
  }
  if (tid < Dh) HFout[(size_t)seq * Dh + tid] = sh[tid];
  if (tid == 0)
    tdm_store_2d_f32(sJ, Jout + (size_t)seq * Dh * Dh, Dh, Dh, Dh);
  __builtin_amdgcn_s_wait_tensorcnt(0);
}

// ---------------------------------------------------------------------------
// block reductions over 128 threads
// ---------------------------------------------------------------------------
__device__ __forceinline__ float blk_sum128(float* red, int i, float v) {
  red[i] = v; __syncthreads();
  for (int s = 64; s > 0; s >>= 1) { if (i < s) red[i] += red[i + s]; __syncthreads(); }
  float r = red[0]; __syncthreads(); return r;
}
__device__ __forceinline__ float blk_min128(float* red, int i, float v) {
  red[i] = v; __syncthreads();
  for (int s = 64; s > 0; s >>= 1) { if (i < s) red[i] = fminf(red[i], red[i + s]); __syncthreads(); }
  float r = red[0]; __syncthreads(); return r;
}
__device__ __forceinline__ float blk_max128(float* red, int i, float v) {
  red[i] = v; __syncthreads();
  for (int s = 64; s > 0; s >>= 1) { if (i < s) red[i] = fmaxf(red[i], red[i + s]); __syncthreads(); }
  float r = red[0]; __syncthreads(); return r;
}

// ---------------------------------------------------------------------------
// K4: row-wise Frobenius / cosine stats for pair (J0, J_{1+k}) + output stats.
// 1 block = (batch, k); thread i handles matrix row i.
// ---------------------------------------------------------------------------
__global__ __launch_bounds__(128)
void stats_pair_kernel(const float* __restrict__ J, const float* __restrict__ HF,
                       float* __restrict__ out) {
  __shared__ float red[128];
  int b = blockIdx.x >> 1;
  int k = blockIdx.x & 1;
  int i = threadIdx.x;
  const float* J0 = J + (size_t)(b * 3) * Dh * Dh + (size_t)i * Dh;
  const float* Jk = J + (size_t)(b * 3 + 1 + k) * Dh * Dh + (size_t)i * Dh;
  float dot = 0.f, n0 = 0.f, nk = 0.f, df = 0.f;
  for (int j = 0; j < Dh; ++j) {
    float a = J0[j], c = Jk[j];
    dot += a * c; n0 += a * a; nk += c * c;
    float d = a - c; df += d * d;
  }
  float n0r = sqrtf(n0), nkr = sqrtf(nk);
  float s0 = (n0r == 0.f) ? 1.f : n0r;
  float sk = (nkr == 0.f) ? 1.f : nkr;
  float g = dot / (s0 * sk);
  float w = n0r * nkr;

  float gmin = blk_min128(red, i, g);
  float gmax = blk_max128(red, i, g);
  float gsum = blk_sum128(red, i, g);
  float wsum = blk_sum128(red, i, w);
  float wg   = blk_sum128(red, i, w * g);
  float frod = blk_sum128(red, i, df);
  float fron = blk_sum128(red, i, n0);

  float h0 = HF[(size_t)(b * 3) * Dh + i];
  float hk = HF[(size_t)(b * 3 + 1 + k) * Dh + i];
  float dh = h0 - hk;
  float fd   = blk_sum128(red, i, dh * dh);
  float hdot = blk_sum128(red, i, h0 * hk);
  float hn0  = blk_sum128(red, i, h0 * h0);
  float hnk  = blk_sum128(red, i, hk * hk);

  if (i == 0) {
    float* o = out + (size_t)b * 21;
    o[1 + k]  = sqrtf(frod);
    o[3 + k]  = gmin;
    o[5 + k]  = gmax;
    o[7 + k]  = gsum * (1.0f / 128.0f);
    o[9 + k]  = wg / ((wsum == 0.f) ? 1.f : wsum);
    o[17 + k] = sqrtf(fd);
    o[19 + k] = hdot / (fmaxf(sqrtf(hn0), 1e-12f) * fmaxf(sqrtf(hnk), 1e-12f));
    if (k == 0) o[0] = sqrtf(fron);
  }
}

// ---------------------------------------------------------------------------
// K5/K7: one-sided Jacobi on columns -> singular values; optionally
// normalize surviving columns in-place (orthonormal range basis U).
// Matrix staged in LDS via TDM; 64 disjoint column pairs/round, 4 lanes/pair.
// ---------------------------------------------------------------------------
__global__ __launch_bounds__(256)
void jacobi_kernel(const float* __restrict__ src, float* __restrict__ sv,
                   float* __restrict__ Uout, int normalize) {
  __shared__ float A[Dh * Dh];
  __shared__ float snorm[Dh];
  int tid = threadIdx.x;
  int pair = tid >> 2, sub = tid & 3;
  if (tid == 0)
    tdm_load_2d_f32(A, src + (size_t)blockIdx.x * Dh * Dh, Dh, Dh, Dh);
  __builtin_amdgcn_s_wait_tensorcnt(0);
  __syncthreads();

  const int NSWEEP = 6;
  for (int sweep = 0; sweep < NSWEEP; ++sweep) {
    for (int r = 0; r < 127; ++r) {
      int p, q;
      if (pair == 0) { p = r % 127; q = 127; }
      else { p = (r + pair) % 127; q = (r + 127 - pair) % 127; }
      if (p > q) { int tmp = p; p = q; q = tmp; }
      float app = 0.f, aqq = 0.f, apq = 0.f;
      int r0 = sub * 32;
      for (int rr = r0; rr < r0 + 32; ++rr) {
        float u = A[rr * Dh + p], v = A[rr * Dh + q];
        app += u * u; aqq += v * v; apq += u * v;
      }
      app += __shfl_xor(app, 1, 32); app += __shfl_xor(app, 2, 32);
      aqq += __shfl_xor(aqq, 1, 32); aqq += __shfl_xor(aqq, 2, 32);
      apq += __shfl_xor(apq, 1, 32); apq += __shfl_xor(apq, 2, 32);
      if (fabsf(apq) > 1e-30f) {
        float zeta = (aqq - app) / (2.0f * apq);
        float t = ((zeta >= 0.f) ? 1.f : -1.f) / (fabsf(zeta) + sqrtf(1.f + zeta * zeta));
        float c = 1.0f / sqrtf(1.f + t * t);
        float s = c * t;
        for (int rr = r0; rr < r0 + 32; ++rr) {
          float u = A[rr * Dh + p], v = A[rr * Dh + q];
          A[rr * Dh + p] = c * u - s * v;
          A[rr * Dh + q] = s * u + c * v;
        }
      }
      __syncthreads();
    }
  }
  // singular values = column norms
  if (tid < Dh) {
    float nn = 0.f;
    for (int rr = 0; rr < Dh; ++rr) { float u = A[rr * Dh + tid]; nn += u * u; }
    float s = sqrtf(nn);
    snorm[tid] = s;
    sv[(size_t)blockIdx.x * Dh + tid] = s;
  }
  __syncthreads();
  if (normalize) {
    if (tid < Dh) {
      float s = snorm[tid];
      float sc = (s > RANK_THRESH) ? (1.0f / s) : 0.0f;  // drop null directions
      for (int rr = 0; rr < Dh; ++rr) A[rr * Dh + tid] *= sc;
    }
    __syncthreads();
    if (tid == 0)
      tdm_store_2d_f32(A, Uout + (size_t)blockIdx.x * Dh * Dh, Dh, Dh, Dh);
    __builtin_amdgcn_s_wait_tensorcnt(0);
  }
}

// ---------------------------------------------------------------------------
// K6: M = U0^T @ Uk  (principal-angle matrix), WMMA f32 16x16x4, TDM staging.
// ---------------------------------------------------------------------------
__global__ __launch_bounds__(256)
void m_kernel(const float* __restrict__ U, float* __restrict__ M) {
  __shared__ float sU0[Dh * Dh];
  __shared__ float sUk[Dh * Dh];
  int tid = threadIdx.x, wv = tid >> 5, lane = tid & 31;
  int b = blockIdx.x >> 1, k = blockIdx.x & 1;
  const float* U0 = U + (size_t)(b * 3) * Dh * Dh;
  const float* Uk = U + (size_t)(b * 3 + 1 + k) * Dh * Dh;
  if (tid == 0) {
    tdm_load_2d_f32(sU0, U0, Dh, Dh, Dh);
    tdm_load_2d_f32(sUk, Uk, Dh, Dh, Dh);
  }
  __builtin_amdgcn_s_wait_tensorcnt(0);
  __syncthreads();

  int ml = lane & 15;
  int khalf = (lane >> 4) * 2;
  int mg = (wv << 4) + ml;
  v8f zero = {0.f, 0.f, 0.f, 0.f, 0.f, 0.f, 0.f, 0.f};
  v8f acc[8];
  for (int q = 0; q < 8; ++q) acc[q] = zero;
  for (int k0 = 0; k0 < Dh; k0 += 4) {
    int kk = k0 + khalf;
    v2f a;
    a[0] = sU0[kk * Dh + mg];        // A[m][k] = U0[k][m] (transposed read)
    a[1] = sU0[(kk + 1) * Dh + mg];
    for (int nt = 0; nt < 8; ++nt) {
      v2f bb;
      int ng = (nt << 4) + ml;
      bb[0] = sUk[kk * Dh + ng];
      bb[1] = sUk[(kk + 1) * Dh + ng];
      acc[nt] = __builtin_amdgcn_wmma_f32_16x16x4_f32(false, a, false, bb, (short)0, acc[nt], false, false);
    }
  }
  float* Mo = M + (size_t)blockIdx.x * Dh * Dh;
  for (int nt = 0; nt < 8; ++nt) {
    for (int r = 0; r < 8; ++r) {
      int mm = (wv << 4) + r + ((lane >> 4) << 3);
      int nn = (nt << 4) + (lane & 15);
      Mo[mm * Dh + nn] = acc[nt][r];
    }
  }
}

// ---------------------------------------------------------------------------
// K8: ranks, soft rank, subspace-similarity stats -> out cols 11..16
// ---------------------------------------------------------------------------
__global__ __launch_bounds__(128)
void final_kernel(const float* __restrict__ SV, const float* __restrict__ MSV,
                  float* __restrict__ out) {
  __shared__ float red[128];
  int b = blockIdx.x, i = threadIdx.x;
  const float sstd = RANK_THRESH / 1.1774100f;   // thresh / sqrt(2 ln 2)
  float s0 = SV[(size_t)(b * 3) * Dh + i];
  float s1 = SV[(size_t)(b * 3 + 1) * Dh + i];
  float s2 = SV[(size_t)(b * 3 + 2) * Dh + i];
  float hr0 = blk_sum128(red, i, (s0 > RANK_THRESH) ? 1.f : 0.f);
  float hr1 = blk_sum128(red, i, (s1 > RANK_THRESH) ? 1.f : 0.f);
  float hr2 = blk_sum128(red, i, (s2 > RANK_THRESH) ? 1.f : 0.f);
  float rat = s0 / sstd;
  float soft = blk_sum128(red, i, __expf(-0.5f * rat * rat));
  float m0 = MSV[(size_t)(b * 2) * Dh + i];
  float m1 = MSV[(size_t)(b * 2 + 1) * Dh + i];
  float mx0 = blk_max128(red, i, m0);
  float mx1 = blk_max128(red, i, m1);
  float sm0 = blk_sum128(red, i, m0);
  float sm1 = blk_sum128(red, i, m1);
  if (i == 0) {
    float* o = out + (size_t)b * 21;
    o[11] = hr0;
    o[12] = 128.0f - soft;
    o[13] = mx0;
    o[14] = mx1;
    o[15] = sm0 / fmaxf(fminf(hr0, hr1), 1.0f);
    o[16] = sm1 / fmaxf(fminf(hr0, hr2), 1.0f);
  }
}

// ---------------------------------------------------------------------------
extern "C" void kernel_launch(void* const* d_in, const int* in_sizes, int n_in,
                              void* d_out, int out_size, void* d_ws, size_t ws_size,
                              hipStream_t stream) {
  (void)in_sizes; (void)n_in; (void)out_size; (void)ws_size;
  const int*   tok = (const int*)d_in[0];
  const float* emb = (const float*)d_in[1];
  const float* Wih = (const float*)d_in[2];
  const float* Whh = (const float*)d_in[3];
  const float* bih = (const float*)d_in[4];
  const float* bhh = (const float*)d_in[5];
  float* out = (float*)d_out;
  float* ws  = (float*)d_ws;

  float* X    = ws + X_OFF;
  float* H    = ws + H_OFF;
  float* HF   = ws + HF_OFF;
  float* Jw   = ws + J_OFF;
  float* SVw  = ws + SV_OFF;
  float* Mw   = ws + M_OFF;
  float* MSVw = ws + MSV_OFF;

  embed_pe_kernel<<<(NSEQ * Lh * Dh + 255) / 256, 256, 0, stream>>>(tok, emb, X);
  prefix_kernel<<<NSEQ / 16, 256, 0, stream>>>(X, Wih, Whh, bih, bhh, H);
  suffix_jac_kernel<<<NSEQ, 256, 0, stream>>>(X, Wih, Whh, bih, bhh, H, HF, Jw);
  stats_pair_kernel<<<Bsz * 2, 128, 0, stream>>>(Jw, HF, out);          // before U overwrite
  jacobi_kernel<<<NSEQ, 256, 0, stream>>>(Jw, SVw, Jw, 1);              // J -> U, s(J)
  m_kernel<<<Bsz * 2, 256, 0, stream>>>(Jw, Mw);                        // M = U0^T Uk
  jacobi_kernel<<<Bsz * 2, 256, 0, stream>>>(Mw, MSVw, Mw, 0);          // s(M)
  final_kernel<<<Bsz, 128, 0, stream>>>(SVw, MSVw, out);
}